// NNPM_69544110457401
// MI455X (gfx1250) — compile-verified
//
#include <hip/hip_runtime.h>
#include <hip/hip_bf16.h>

typedef __attribute__((ext_vector_type(2))) float v2f;
typedef __attribute__((ext_vector_type(8))) float v8f;

#define D_FEAT 128
#define WT_PITCH 132  // 128 + 4 pad: lane bank = (4n + k + kofs) % 64 -> all 64 banks, conflict-free

// Phase 1: P[n, 0:2] = h[n] @ W[0:128, :], P[n, 2:4] = h[n] @ W[128:256, :]
// One wave computes a 16-node tile via V_WMMA_F32_16X16X4_F32 chained over K=128.
// B operand (zero-padded 128x16 weight matrix) staged transposed in LDS so the
// inner loop is branch-free: global_load_b64 (A) + ds_load_b64 (B) + wmma.
__global__ void node_proj_wmma(const float* __restrict__ h,
                               const float* __restrict__ W,
                               float* __restrict__ P,
                               int n_nodes) {
  __shared__ float Wt[16 * WT_PITCH];  // Wt[n*WT_PITCH + k] = W2[k][n], 8.25 KB

  // Cooperative fill: W2[k][n] = (n<2) ? W[k][n] : (n<4 ? W[128+k][n-2] : 0)
  for (int i = threadIdx.x; i < 16 * D_FEAT; i += blockDim.x) {
    const int n = i >> 7;       // 0..15
    const int k = i & (D_FEAT - 1);
    float v = 0.0f;
    if (n < 2)      v = W[(size_t)k * 2 + n];
    else if (n < 4) v = W[(size_t)(D_FEAT + k) * 2 + (n - 2)];
    Wt[n * WT_PITCH + k] = v;
  }
  __syncthreads();

  const int lane = threadIdx.x & 31;
  const int wave = (blockIdx.x * blockDim.x + threadIdx.x) >> 5;
  const int row_base = wave * 16;
  if (row_base >= n_nodes) return;  // wave-uniform: EXEC stays all-1s for WMMA

  const int m     = lane & 15;   // A: row M in tile; B/D: column N
  const int khalf = lane >> 4;   // 0 -> K{0,1}, 1 -> K{2,3} (32-bit A/B layout)
  const int kofs  = khalf * 2;

  const float* __restrict__ hrow    = h + (size_t)(row_base + m) * D_FEAT + kofs;
  const float* __restrict__ wt_lane = &Wt[m * WT_PITCH + kofs];

  v8f acc = {};
#pragma unroll 4
  for (int k = 0; k < D_FEAT; k += 4) {
    const v2f a  = *(const v2f*)(hrow + k);     // global_load_b64, unconditional
    const v2f bv = *(const v2f*)(wt_lane + k);  // ds_load_b64, conflict-free
    // D = A(16x4,f32) x B(4x16,f32) + C ; emits v_wmma_f32_16x16x4_f32
    acc = __builtin_amdgcn_wmma_f32_16x16x4_f32(false, a, false, bv,
                                                (short)0, acc, false, false);
  }

  // C/D layout: VGPR j, lanes 0-15 -> M=j, N=lane; lanes 16-31 -> M=j+8, N=lane-16.
  // Only N<4 columns carry real outputs.
  if (m < 4) {
    const int mhi = khalf * 8;
#pragma unroll
    for (int j = 0; j < 8; ++j) {
      P[(size_t)(row_base + mhi + j) * 4 + m] = acc[j];
    }
  }
}

// Phase 2: per edge, logits = w * (P[src][0:2] + P[dst][2:4]) + b; sigmoid; softmax(2).
__global__ void edge_predict(const float* __restrict__ P,
                             const float* __restrict__ w,
                             const int* __restrict__ src,
                             const int* __restrict__ dst,
                             const float* __restrict__ b,
                             float* __restrict__ out,
                             int n_edges) {
  const int e = blockIdx.x * blockDim.x + threadIdx.x;
  if (e >= n_edges) return;

  const int   s  = src[e];
  const int   d  = dst[e];
  const float we = w[e];

  const float2 ps = *(const float2*)(P + (size_t)s * 4);      // src @ W_top  (cols 0,1)
  const float2 pd = *(const float2*)(P + (size_t)d * 4 + 2);  // dst @ W_bot  (cols 0,1)

  const float l0 = we * (ps.x + pd.x) + b[0];
  const float l1 = we * (ps.y + pd.y) + b[1];

  const float s0 = 1.0f / (1.0f + __expf(-l0));
  const float s1 = 1.0f / (1.0f + __expf(-l1));

  // 2-class softmax, max-subtracted for stability
  const float mx  = fmaxf(s0, s1);
  const float e0  = __expf(s0 - mx);
  const float e1  = __expf(s1 - mx);
  const float inv = 1.0f / (e0 + e1);

  float2 r;
  r.x = e0 * inv;
  r.y = e1 * inv;
  *(float2*)(out + (size_t)e * 2) = r;
}

extern "C" void kernel_launch(void* const* d_in, const int* in_sizes, int n_in,
                              void* d_out, int out_size, void* d_ws, size_t ws_size,
                              hipStream_t stream) {
  const float* h   = (const float*)d_in[0];
  const float* w   = (const float*)d_in[1];
  const int*   src = (const int*)d_in[2];
  const int*   dst = (const int*)d_in[3];
  const float* W   = (const float*)d_in[4];
  const float* b   = (const float*)d_in[5];
  float*       out = (float*)d_out;

  const int n_nodes = in_sizes[0] / D_FEAT;
  const int n_edges = in_sizes[2];

  float* P = (float*)d_ws;  // n_nodes * 4 floats = 800 KB (L2-resident)

  // Phase 1: one 16-row tile per wave, 8 waves per block
  const int waves   = (n_nodes + 15) / 16;
  const int threads = waves * 32;
  const int block1  = 256;
  const int grid1   = (threads + block1 - 1) / block1;
  node_proj_wmma<<<grid1, block1, 0, stream>>>(h, W, P, n_nodes);

  // Phase 2: one thread per edge
  const int block2 = 256;
  const int grid2  = (n_edges + block2 - 1) / block2;
  edge_predict<<<grid2, block2, 0, stream>>>(P, w, src, dst, b, out, n_edges);
}